// UserItemWithRepostTrimmedTimeEmbeddingWithExtraEdge_70755291234323
// MI455X (gfx1250) — compile-verified
//
#include <hip/hip_runtime.h>

// Problem constants (match reference)
#define D1 64     // NINP
#define D2 128    // 2*NINP

typedef __attribute__((ext_vector_type(2))) float v2f;
typedef __attribute__((ext_vector_type(8))) float v8f;

// ---------------------------------------------------------------------------
// Degree: deg[i] = 1 (self loop) + #edges with col==i
// ---------------------------------------------------------------------------
__global__ void k_deg_init(int* __restrict__ deg, int N) {
    int i = blockIdx.x * blockDim.x + threadIdx.x;
    if (i < N) deg[i] = 1;
}

__global__ void k_deg_count(const int* __restrict__ col, int* __restrict__ deg, int E) {
    int e = blockIdx.x * blockDim.x + threadIdx.x;
    if (e < E) atomicAdd(&deg[col[e]], 1);
}

__global__ void k_dinv(const int* __restrict__ deg, float* __restrict__ dinv, int N) {
    int i = blockIdx.x * blockDim.x + threadIdx.x;
    if (i < N) dinv[i] = rsqrtf(fmaxf((float)deg[i], 1.0f));
}

// ---------------------------------------------------------------------------
// FP32 WMMA GEMM: C[M,NOUT] = A[M,K] * B[K,NOUT], row-major.
// One wave computes a 16x16 f32 tile via V_WMMA_F32_16X16X4_F32, K stepped 4.
// blockDim.x = (NOUT/16)*32 waves -> block covers 16 rows x NOUT cols.
// A 16x4 layout: lanes 0-15 hold rows 0-15 with K=0,1 in v[0],v[1];
//                lanes 16-31 hold K=2,3.  B 4x16 is the mirror layout.
// C/D: VGPR i -> M = i + (lane<16 ? 0 : 8), N = lane&15.
// ---------------------------------------------------------------------------
template <int K, int NOUT>
__global__ void k_gemm_wmma(const float* __restrict__ A, const float* __restrict__ B,
                            float* __restrict__ C) {
    const int wave  = threadIdx.x >> 5;
    const int lane  = threadIdx.x & 31;
    const int rowb  = blockIdx.x * 16;
    const int colb  = wave * 16;
    const int mr    = lane & 15;           // A row within tile
    const int khalf = (lane >> 4) << 1;    // 0 or 2: which K pair this lane holds
    const int n     = lane & 15;           // B/C column within tile

    v8f acc = {};
    const float* ap  = A + (size_t)(rowb + mr) * K + khalf;
    const float* bp0 = B + (size_t)khalf * NOUT + colb + n;

#pragma unroll 4
    for (int k = 0; k < K; k += 4) {
        v2f a, b;
        a.x = ap[k];
        a.y = ap[k + 1];
        const float* bp = bp0 + (size_t)k * NOUT;
        b.x = bp[0];
        b.y = bp[NOUT];
        acc = __builtin_amdgcn_wmma_f32_16x16x4_f32(
            /*neg_a=*/false, a, /*neg_b=*/false, b,
            /*c_mod=*/(short)0, acc, /*reuse_a=*/false, /*reuse_b=*/false);
    }

    const int mo = (lane >> 4) << 3;       // 0 or 8
    float* cp = C + (size_t)(rowb + mo) * NOUT + colb + n;
#pragma unroll
    for (int i = 0; i < 8; ++i) cp[(size_t)i * NOUT] = acc[i];
}

// ---------------------------------------------------------------------------
// agg[i][j] = bias[j] + h[i][j] * dinv[i]^2   (bias + self-loop message fused)
// ---------------------------------------------------------------------------
template <int F>
__global__ void k_agg_init(const float* __restrict__ h, const float* __restrict__ bias,
                           const float* __restrict__ dinv, float* __restrict__ agg,
                           int N) {
    long long idx = (long long)blockIdx.x * blockDim.x + threadIdx.x;
    if (idx >= (long long)N * F) return;
    int i = (int)(idx / F);
    int j = (int)(idx % F);
    float d = dinv[i];
    agg[idx] = bias[j] + h[idx] * d * d;
}

// ---------------------------------------------------------------------------
// Edge scatter: agg[col[e]][:] += h[row[e]][:] * dinv[row]*dinv[col]
// LPE lanes per edge, each lane handles 4 consecutive floats (float4 load,
// 4x global_atomic_add_f32). agg lives (mostly) in the 192MB L2, so the
// atomic RMWs resolve at L2 bandwidth.
// ---------------------------------------------------------------------------
template <int F, int LPE>   // F = LPE*4
__global__ void k_scatter(const int* __restrict__ row, const int* __restrict__ col,
                          const float* __restrict__ dinv, const float* __restrict__ h,
                          float* __restrict__ agg, int E) {
    long long t = (long long)blockIdx.x * blockDim.x + threadIdx.x;
    long long e = t / LPE;
    if (e >= E) return;
    int j = (int)(t % LPE) * 4;
    int r = row[e];
    int c = col[e];
    float norm = dinv[r] * dinv[c];
    const float4 hv = *(const float4*)(h + (size_t)r * F + j);
    float* dst = agg + (size_t)c * F + j;
    unsafeAtomicAdd(dst + 0, hv.x * norm);
    unsafeAtomicAdd(dst + 1, hv.y * norm);
    unsafeAtomicAdd(dst + 2, hv.z * norm);
    unsafeAtomicAdd(dst + 3, hv.w * norm);
}

// ---------------------------------------------------------------------------
// out[t][:] = agg2[input[t]][:]
// ---------------------------------------------------------------------------
__global__ void k_gather(const int* __restrict__ input, const float* __restrict__ agg2,
                         float* __restrict__ out, int T) {
    long long idx = (long long)blockIdx.x * blockDim.x + threadIdx.x;
    if (idx >= (long long)T * D1) return;
    int t = (int)(idx / D1);
    int j = (int)(idx % D1);
    out[idx] = agg2[(size_t)input[t] * D1 + j];
}

// ---------------------------------------------------------------------------
extern "C" void kernel_launch(void* const* d_in, const int* in_sizes, int n_in,
                              void* d_out, int out_size, void* d_ws, size_t ws_size,
                              hipStream_t stream) {
    const int*   input = (const int*)d_in[0];
    // d_in[1] input_timestamp, d_in[2] input_id: unused by reference forward
    const int*   edge  = (const int*)d_in[3];
    const float* emb   = (const float*)d_in[4];
    const float* W1    = (const float*)d_in[5];
    const float* b1    = (const float*)d_in[6];
    const float* W2    = (const float*)d_in[7];
    const float* b2    = (const float*)d_in[8];
    float*       out   = (float*)d_out;

    const int N = in_sizes[4] / D1;  // 200000 nodes (divisible by 16)
    const int E = in_sizes[3] / 2;   // 2,000,000 edges
    const int T = in_sizes[0];       // 12800 = B*L

    const int* row = edge;
    const int* col = edge + E;

    // Workspace layout (all 16B-aligned; total ~309 MB)
    char* ws = (char*)d_ws;
    int*   deg  = (int*)ws;          ws += (size_t)N * 4;
    float* dinv = (float*)ws;        ws += (size_t)N * 4;
    float* h1   = (float*)ws;        ws += (size_t)N * D2 * 4;   // emb @ W1
    float* agg1 = (float*)ws;        ws += (size_t)N * D2 * 4;   // layer-1 aggregate
    float* h2   = (float*)ws;        ws += (size_t)N * D1 * 4;   // agg1 @ W2
    float* agg2 = (float*)ws;                                    // layer-2 aggregate

    // 1. degrees + dinv
    k_deg_init<<<(N + 255) / 256, 256, 0, stream>>>(deg, N);
    k_deg_count<<<(E + 255) / 256, 256, 0, stream>>>(col, deg, E);
    k_dinv<<<(N + 255) / 256, 256, 0, stream>>>(deg, dinv, N);

    // 2. h1 = emb @ W1  (fp32 WMMA; 8 waves/block = 16 rows x 128 cols)
    k_gemm_wmma<D1, D2><<<N / 16, 256, 0, stream>>>(emb, W1, h1);

    // 3. agg1 = b1 + self-loop, then edge scatter (32 lanes/edge x float4)
    {
        long long n1 = (long long)N * D2;
        k_agg_init<D2><<<(unsigned)((n1 + 255) / 256), 256, 0, stream>>>(h1, b1, dinv, agg1, N);
        long long w1 = (long long)E * (D2 / 4);
        k_scatter<D2, D2 / 4><<<(unsigned)((w1 + 255) / 256), 256, 0, stream>>>(row, col, dinv, h1, agg1, E);
    }

    // 4. h2 = agg1 @ W2  (4 waves/block = 16 rows x 64 cols)
    k_gemm_wmma<D2, D1><<<N / 16, 128, 0, stream>>>(agg1, W2, h2);

    // 5. agg2 = b2 + self-loop, then edge scatter (16 lanes/edge x float4)
    {
        long long n2 = (long long)N * D1;
        k_agg_init<D1><<<(unsigned)((n2 + 255) / 256), 256, 0, stream>>>(h2, b2, dinv, agg2, N);
        long long w2 = (long long)E * (D1 / 4);
        k_scatter<D1, D1 / 4><<<(unsigned)((w2 + 255) / 256), 256, 0, stream>>>(row, col, dinv, h2, agg2, E);
    }

    // 6. out = agg2[input]
    {
        long long ng = (long long)T * D1;
        k_gather<<<(unsigned)((ng + 255) / 256), 256, 0, stream>>>(input, agg2, out, T);
    }
}